// ResidualVectorQuantizer_80461917323715
// MI455X (gfx1250) — compile-verified
//
#include <hip/hip_runtime.h>

typedef __attribute__((ext_vector_type(2))) float v2f;
typedef __attribute__((ext_vector_type(8))) float v8f;

#define BETA   0.25f
#define DIM    32
#define NLAY   4
#define NE     256
#define NBLK   16      // 256 entries / 16 columns per WMMA
#define KSTEPS 8       // 32 dims / K=4 per WMMA

// Dynamic LDS layout (bytes):
//   [0, 131072)        blayout float2[NLAY][NBLK][KSTEPS][32]  (B-operand lane layout)
//   [131072, 135168)   hbuf    float[NLAY*NE]       (0.5*||c||^2)
//   [135168, 137216)   wsK     u64[8][32]           (per-wave winner keys, 2 tiles)
//   [137216, 138240)   lred    float[256]           (loss block reduction)
#define LDS_BYTES 138240

__global__ void rvq_fused_kernel(const float* __restrict__ x,
                                 const float* __restrict__ cb,
                                 float* __restrict__ out,
                                 float* __restrict__ partials,
                                 int N) {
  extern __shared__ char smem[];
  float2*             ldsB = (float2*)smem;
  float*              hbuf = (float*)(smem + 131072);
  unsigned long long* wsK  = (unsigned long long*)(smem + 135168);
  float*              lred = (float*)(smem + 137216);

  const int tid = threadIdx.x;

  // ---- Prologue: build B-operand layout + half-norms (codebooks hit L2) ----
  for (int t = tid; t < NLAY * NBLK * KSTEPS * 32; t += blockDim.x) {
    int q  = t & 31;
    int kk = (t >> 5) & 7;
    int nb = (t >> 8) & 15;
    int l  = (t >> 12);
    int e  = nb * 16 + (q & 15);
    int hh = q >> 4;
    const float* src = cb + ((l * NE + e) * DIM + 4 * kk + 2 * hh);
    ldsB[t] = make_float2(src[0], src[1]);
  }
  for (int i = tid; i < NLAY * NE; i += blockDim.x) {
    const float* c = cb + (size_t)i * DIM;
    float s = 0.f;
    #pragma unroll
    for (int d = 0; d < DIM; ++d) s += c[d] * c[d];
    hbuf[i] = 0.5f * s;
  }
  __syncthreads();

  const int lane = tid & 31;
  const int wave = tid >> 5;
  const int h    = lane >> 4;   // which K-half this lane carries
  const int m    = lane & 15;   // row (A) / column (B,C,D) index within tile
  unsigned long long* wsW = wsK + wave * 32;

  const int numPairs = N / 32;                 // two 16-row tiles per wave
  const int waveGlob = blockIdx.x * (blockDim.x >> 5) + wave;
  const int waveTot  = gridDim.x * (blockDim.x >> 5);

  const size_t lossOff = (size_t)N * DIM;      // loss at out[N*D]
  const size_t idxOff  = lossOff + 1;          // indices after loss

  float lossAcc = 0.f;
  const float2* lb = ldsB + lane;

  for (int pair = waveGlob; pair < numPairs; pair += waveTot) {
    const int row0 = pair * 32 + m;
    const int row1 = row0 + 16;
    float rr0[16], qa0[16], rr1[16], qa1[16];
    #pragma unroll
    for (int kk = 0; kk < KSTEPS; ++kk) {      // load x tiles in A layout
      float2 p0 = *(const float2*)(x + (size_t)row0 * DIM + 4 * kk + 2 * h);
      float2 p1 = *(const float2*)(x + (size_t)row1 * DIM + 4 * kk + 2 * h);
      rr0[2 * kk] = p0.x; rr0[2 * kk + 1] = p0.y;
      rr1[2 * kk] = p1.x; rr1[2 * kk + 1] = p1.y;
      qa0[2 * kk] = 0.f; qa0[2 * kk + 1] = 0.f;
      qa1[2 * kk] = 0.f; qa1[2 * kk + 1] = 0.f;
    }

    #pragma unroll 1                           // keep rolled: I-cache locality
    for (int l = 0; l < NLAY; ++l) {
      float best0[8], bidx0[8], best1[8], bidx1[8];
      #pragma unroll
      for (int v = 0; v < 8; ++v) {
        best0[v] = -3.4e38f; bidx0[v] = 0.f;
        best1[v] = -3.4e38f; bidx1[v] = 0.f;
      }

      // ---- double-buffered B operands; each fetch feeds 16 WMMAs (2 tiles) ----
      float2 bc[KSTEPS];
      #pragma unroll
      for (int kk = 0; kk < KSTEPS; ++kk)
        bc[kk] = lb[((size_t)(l * NBLK) * KSTEPS + kk) * 32];

      #pragma unroll 2
      for (int nb = 0; nb < NBLK; ++nb) {
        float2 bn[KSTEPS];
        const int nofs = (l * NBLK + ((nb + 1) & 15)) * KSTEPS;
        #pragma unroll
        for (int kk = 0; kk < KSTEPS; ++kk)
          bn[kk] = lb[((size_t)nofs + kk) * 32];

        float hn = hbuf[l * NE + nb * 16 + m];
        v8f acc0, acc1;
        #pragma unroll
        for (int v = 0; v < 8; ++v) { acc0[v] = -hn; acc1[v] = -hn; }
        #pragma unroll
        for (int kk = 0; kk < KSTEPS; ++kk) {
          v2f b;  b[0]  = bc[kk].x;       b[1]  = bc[kk].y;
          v2f a0; a0[0] = rr0[2 * kk];    a0[1] = rr0[2 * kk + 1];
          v2f a1; a1[0] = rr1[2 * kk];    a1[1] = rr1[2 * kk + 1];
          acc0 = __builtin_amdgcn_wmma_f32_16x16x4_f32(
              false, a0, false, b, (short)0, acc0, false, false);
          acc1 = __builtin_amdgcn_wmma_f32_16x16x4_f32(
              false, a1, false, b, (short)0, acc1, false, false);
        }
        float cf = (float)(nb * 16 + m);
        #pragma unroll
        for (int v = 0; v < 8; ++v) {       // strict > : smallest col wins ties per lane
          bool g0 = acc0[v] > best0[v];
          best0[v] = g0 ? acc0[v] : best0[v];
          bidx0[v] = g0 ? cf : bidx0[v];
          bool g1 = acc1[v] > best1[v];
          best1[v] = g1 ? acc1[v] : best1[v];
          bidx1[v] = g1 ? cf : bidx1[v];
        }
        #pragma unroll
        for (int kk = 0; kk < KSTEPS; ++kk) bc[kk] = bn[kk];
      }

      // ---- pack (score, col) into sortable u64 keys: branchless argmax ----
      unsigned long long key0[8], key1[8];
      #pragma unroll
      for (int v = 0; v < 8; ++v) {
        int f0 = __float_as_int(best0[v]);
        unsigned u0 = (unsigned)f0 ^ (unsigned)((f0 >> 31) | 0x80000000);
        key0[v] = ((unsigned long long)u0 << 8) | (unsigned)(255 - (int)bidx0[v]);
        int f1 = __float_as_int(best1[v]);
        unsigned u1 = (unsigned)f1 ^ (unsigned)((f1 >> 31) | 0x80000000);
        key1[v] = ((unsigned long long)u1 << 8) | (unsigned)(255 - (int)bidx1[v]);
      }
      #pragma unroll
      for (int off = 1; off < 16; off <<= 1) {
        #pragma unroll
        for (int v = 0; v < 8; ++v) {
          unsigned long long o0 = __shfl_xor(key0[v], off);
          key0[v] = (o0 > key0[v]) ? o0 : key0[v];   // u64 max, branchless
          unsigned long long o1 = __shfl_xor(key1[v], off);
          key1[v] = (o1 > key1[v]) ? o1 : key1[v];
        }
      }
      // deposit winners (DS in-order per wave), then read row m's winner
      #pragma unroll
      for (int v = 0; v < 8; ++v)
        if (m == v) { wsW[v + 8 * h] = key0[v]; wsW[16 + v + 8 * h] = key1[v]; }
      __asm__ volatile("s_wait_dscnt 0x0" ::: "memory");
      unsigned long long kr0 = wsW[m];
      unsigned long long kr1 = wsW[16 + m];
      int e0 = 255 - (int)(kr0 & 0xff);
      int e1 = 255 - (int)(kr1 & 0xff);
      unsigned w0 = (unsigned)(kr0 >> 8);
      unsigned w1 = (unsigned)(kr1 >> 8);
      int g0 = (w0 & 0x80000000u) ? (int)(w0 ^ 0x80000000u) : (int)~w0;
      int g1 = (w1 & 0x80000000u) ? (int)(w1 ^ 0x80000000u) : (int)~w1;
      float smax0 = __int_as_float(g0);
      float smax1 = __int_as_float(g1);

      // loss: dist = ||r||^2 - 2*smax
      float rh0 = 0.f, rh1 = 0.f;
      #pragma unroll
      for (int i = 0; i < 16; ++i) { rh0 += rr0[i] * rr0[i]; rh1 += rr1[i] * rr1[i]; }
      float rq0 = rh0 + __shfl_xor(rh0, 16);
      float rq1 = rh1 + __shfl_xor(rh1, 16);
      if (lane < 16) {
        lossAcc += (rq0 - 2.f * smax0) + (rq1 - 2.f * smax1);
        out[idxOff + (size_t)row0 * NLAY + l] = (float)e0;
        out[idxOff + (size_t)row1 * NLAY + l] = (float)e1;
      }

      // gather winning codewords from B layout; residual update + x_q accumulate
      const float2* gp0 = ldsB + ((size_t)(l * NBLK + (e0 >> 4)) * KSTEPS) * 32
                               + ((e0 & 15) + 16 * h);
      const float2* gp1 = ldsB + ((size_t)(l * NBLK + (e1 >> 4)) * KSTEPS) * 32
                               + ((e1 & 15) + 16 * h);
      #pragma unroll
      for (int kk = 0; kk < KSTEPS; ++kk) {
        float2 p0 = gp0[kk * 32];
        float2 p1 = gp1[kk * 32];
        rr0[2 * kk] -= p0.x; rr0[2 * kk + 1] -= p0.y;
        qa0[2 * kk] += p0.x; qa0[2 * kk + 1] += p0.y;
        rr1[2 * kk] -= p1.x; rr1[2 * kk + 1] -= p1.y;
        qa1[2 * kk] += p1.x; qa1[2 * kk + 1] += p1.y;
      }
    }

    #pragma unroll
    for (int kk = 0; kk < KSTEPS; ++kk) {
      *(float2*)(out + (size_t)row0 * DIM + 4 * kk + 2 * h) =
          make_float2(qa0[2 * kk], qa0[2 * kk + 1]);
      *(float2*)(out + (size_t)row1 * DIM + 4 * kk + 2 * h) =
          make_float2(qa1[2 * kk], qa1[2 * kk + 1]);
    }
  }

  // ---- deterministic per-block loss partial ----
  __syncthreads();
  lred[tid] = lossAcc;
  __syncthreads();
  for (int off = 128; off > 0; off >>= 1) {
    if (tid < off) lred[tid] += lred[tid + off];
    __syncthreads();
  }
  if (tid == 0) partials[blockIdx.x] = lred[0];
}

__global__ void rvq_loss_finalize(const float* __restrict__ partials, int n,
                                  float* __restrict__ out, size_t off, float scale) {
  __shared__ float s[256];
  float v = 0.f;
  for (int i = threadIdx.x; i < n; i += 256) v += partials[i];
  s[threadIdx.x] = v;
  __syncthreads();
  for (int o = 128; o > 0; o >>= 1) {
    if ((int)threadIdx.x < o) s[threadIdx.x] += s[threadIdx.x + o];
    __syncthreads();
  }
  if (threadIdx.x == 0) out[off] = s[0] * scale;
}

extern "C" void kernel_launch(void* const* d_in, const int* in_sizes, int n_in,
                              void* d_out, int out_size, void* d_ws, size_t ws_size,
                              hipStream_t stream) {
  const float* x  = (const float*)d_in[0];
  const float* cb = (const float*)d_in[1];
  float* out      = (float*)d_out;
  float* partials = (float*)d_ws;

  const int N = in_sizes[0] / DIM;   // 131072
  const int blocks = 512;            // 4096 waves -> 1 row-pair (32 rows) each

  hipFuncSetAttribute((const void*)rvq_fused_kernel,
                      hipFuncAttributeMaxDynamicSharedMemorySize, LDS_BYTES);

  rvq_fused_kernel<<<blocks, 256, LDS_BYTES, stream>>>(x, cb, out, partials, N);

  const float scale = (1.f + BETA) / ((float)NLAY * (float)N * (float)DIM);
  rvq_loss_finalize<<<1, 256, 0, stream>>>(partials, blocks, out,
                                           (size_t)N * DIM, scale);
}